// KEPCE_GAT_PNA_41566693490868
// MI455X (gfx1250) — compile-verified
//
#include <hip/hip_runtime.h>
#include <hip/hip_bf16.h>
#include <math.h>

typedef __attribute__((ext_vector_type(2))) float v2f;
typedef __attribute__((ext_vector_type(8))) float v8f;

#define NEG_SLOPE 0.2f

// ---- float <-> orderable-uint (monotonic) for hardware atomic min/max ----
__device__ __forceinline__ unsigned f2ord(float f) {
  unsigned u = __float_as_uint(f);
  return (u & 0x80000000u) ? ~u : (u | 0x80000000u);
}
__device__ __forceinline__ float ord2f(unsigned o) {
  unsigned u = (o & 0x80000000u) ? (o ^ 0x80000000u) : ~o;
  return __uint_as_float(u);
}
__device__ __forceinline__ void atomAddF(float* p, float v) {
  __hip_atomic_fetch_add(p, v, __ATOMIC_RELAXED, __HIP_MEMORY_SCOPE_AGENT);
}

// ---------------- generic fill ----------------
__global__ void k_fill_u32(unsigned* __restrict__ p, unsigned v, size_t n) {
  size_t i = (size_t)blockIdx.x * blockDim.x + threadIdx.x;
  size_t stride = (size_t)gridDim.x * blockDim.x;
  for (; i < n; i += stride) p[i] = v;
}

// ---------------- PNA avg_log scalar ----------------
__global__ void k_avglog(const float* __restrict__ deg, float* __restrict__ out) {
  __shared__ float s1[64], s2[64];
  int t = threadIdx.x;
  float d = deg[t];
  s1[t] = logf((float)t + 1.0f) * d;
  s2[t] = d;
  __syncthreads();
  for (int s = 32; s > 0; s >>= 1) {
    if (t < s) { s1[t] += s1[t + s]; s2[t] += s2[t + s]; }
    __syncthreads();
  }
  if (t == 0) out[0] = s1[0] / s2[0];
}

// ---------------- PNA edge pass: pre-net + segment stats via atomics ----------------
__global__ void k_pna_edge(const int* __restrict__ ei, const float* __restrict__ ew,
                           const float* __restrict__ ce, const float* __restrict__ x,
                           const float* __restrict__ We, const float* __restrict__ be,
                           const float* __restrict__ Wpre, const float* __restrict__ bpre,
                           float* __restrict__ cnt, float* __restrict__ sum,
                           float* __restrict__ sumsq, unsigned* __restrict__ minb,
                           unsigned* __restrict__ maxb, long long E) {
  long long e = (long long)blockIdx.x * blockDim.x + threadIdx.x;
  if (e >= E) return;
  int s = ei[e], d = ei[E + e];
  float w = ew[e], c = ce[e];
  float enc[5], xs[5], xd[5], h[5];
#pragma unroll
  for (int j = 0; j < 5; ++j) enc[j] = w * We[j] + c * We[5 + j] + be[j];
#pragma unroll
  for (int i = 0; i < 5; ++i) { xs[i] = x[(long long)s * 5 + i]; xd[i] = x[(long long)d * 5 + i]; }
#pragma unroll
  for (int j = 0; j < 5; ++j) {
    float a = bpre[j];
#pragma unroll
    for (int i = 0; i < 5; ++i) a += xd[i] * Wpre[i * 5 + j];
#pragma unroll
    for (int i = 0; i < 5; ++i) a += xs[i] * Wpre[(5 + i) * 5 + j];
#pragma unroll
    for (int i = 0; i < 5; ++i) a += enc[i] * Wpre[(10 + i) * 5 + j];
    h[j] = a;
  }
  atomAddF(&cnt[d], 1.0f);
#pragma unroll
  for (int j = 0; j < 5; ++j) {
    long long k = (long long)d * 5 + j;
    atomAddF(&sum[k], h[j]);
    atomAddF(&sumsq[k], h[j] * h[j]);
    unsigned o = f2ord(h[j]);
    atomicMin(&minb[k], o);
    atomicMax(&maxb[k], o);
  }
}

// ---------------- PNA node: aggregators + scalers + post/lin ----------------
__global__ void k_pna_node(const float* __restrict__ x, const float* __restrict__ cnt,
                           const float* __restrict__ sum, const float* __restrict__ sumsq,
                           const unsigned* __restrict__ minb, const unsigned* __restrict__ maxb,
                           const float* __restrict__ avg,
                           const float* __restrict__ Wpost, const float* __restrict__ bpost,
                           const float* __restrict__ Wlin, const float* __restrict__ blin,
                           float* __restrict__ xn0, int N) {
  int n = blockIdx.x * blockDim.x + threadIdx.x;
  if (n >= N) return;
  float c = cnt[n];
  float cc = fmaxf(c, 1.0f);
  float logd = logf(cc + 1.0f);
  float al = avg[0];
  float s1 = logd / al, s2 = al / logd;
  float feat[65];
#pragma unroll
  for (int i = 0; i < 5; ++i) feat[i] = x[(long long)n * 5 + i];
  float aggr[20];
#pragma unroll
  for (int j = 0; j < 5; ++j) {
    long long k = (long long)n * 5 + j;
    float mean = sum[k] / cc;
    float msq = sumsq[k] / cc;
    float var = msq - mean * mean;
    float stdv = sqrtf(fmaxf(var, 0.0f) + 1e-5f);
    float mn = (c == 0.0f) ? 0.0f : ord2f(minb[k]);
    float mx = (c == 0.0f) ? 0.0f : ord2f(maxb[k]);
    aggr[j] = mean; aggr[5 + j] = mn; aggr[10 + j] = mx; aggr[15 + j] = stdv;
  }
#pragma unroll
  for (int a = 0; a < 20; ++a) {
    feat[5 + a] = aggr[a];
    feat[25 + a] = aggr[a] * s1;
    feat[45 + a] = aggr[a] * s2;
  }
  float t8[8];
#pragma unroll
  for (int j = 0; j < 8; ++j) {
    float acc = bpost[j];
    for (int i = 0; i < 65; ++i) acc += feat[i] * Wpost[i * 8 + j];
    t8[j] = acc;
  }
#pragma unroll
  for (int j2 = 0; j2 < 8; ++j2) {
    float acc = blin[j2];
#pragma unroll
    for (int j = 0; j < 8; ++j) acc += t8[j] * Wlin[j * 8 + j2];
    xn0[(long long)n * 8 + j2] = acc;
  }
}

// ---------------- GATv2: node-side transforms ----------------
__global__ void k_gat_node(const float* __restrict__ xin, int IN, int OUT,
                           const float* __restrict__ Wl, const float* __restrict__ bl,
                           const float* __restrict__ Wr, const float* __restrict__ br,
                           float* __restrict__ xl, float* __restrict__ xr, int N) {
  int n = blockIdx.x * blockDim.x + threadIdx.x;
  if (n >= N) return;
  float v[32];
  for (int i = 0; i < IN; ++i) v[i] = xin[(long long)n * IN + i];
  for (int j = 0; j < OUT; ++j) {
    float al = bl[j], ar = br[j];
    for (int i = 0; i < IN; ++i) { al += v[i] * Wl[i * OUT + j]; ar += v[i] * Wr[i * OUT + j]; }
    xl[(long long)n * OUT + j] = al;
    xr[(long long)n * OUT + j] = ar;
  }
}

// ---------------- GATv2: edge pass A (segment max of alpha) ----------------
__global__ void k_gat_edge_a(const int* __restrict__ ei, const float* __restrict__ ew,
                             const float* __restrict__ ce, const float* __restrict__ xl,
                             const float* __restrict__ xr, const float* __restrict__ We,
                             const float* __restrict__ att, unsigned* __restrict__ amax,
                             long long E, int H, int C) {
  long long e = (long long)blockIdx.x * blockDim.x + threadIdx.x;
  if (e >= E) return;
  int s = ei[e], d = ei[E + e];
  float w = ew[e], cc = ce[e];
  int HC = H * C;
  for (int h = 0; h < H; ++h) {
    float al = 0.0f;
    for (int c = 0; c < C; ++c) {
      int hc = h * C + c;
      float m = xl[(long long)s * HC + hc] + xr[(long long)d * HC + hc] + (w * We[hc] + cc * We[HC + hc]);
      m = m > 0.0f ? m : NEG_SLOPE * m;
      al += m * att[hc];
    }
    atomicMax(&amax[(long long)d * H + h], f2ord(al));
  }
}

// ---------------- GATv2: edge pass B (exp, denom, weighted numerator) ----------------
__global__ void k_gat_edge_b(const int* __restrict__ ei, const float* __restrict__ ew,
                             const float* __restrict__ ce, const float* __restrict__ xl,
                             const float* __restrict__ xr, const float* __restrict__ We,
                             const float* __restrict__ att, const unsigned* __restrict__ amax,
                             float* __restrict__ denom, float* __restrict__ acc,
                             long long E, int H, int C) {
  long long e = (long long)blockIdx.x * blockDim.x + threadIdx.x;
  if (e >= E) return;
  int s = ei[e], d = ei[E + e];
  float w = ew[e], cc = ce[e];
  int HC = H * C;
  for (int h = 0; h < H; ++h) {
    float al = 0.0f;
    for (int c = 0; c < C; ++c) {
      int hc = h * C + c;
      float m = xl[(long long)s * HC + hc] + xr[(long long)d * HC + hc] + (w * We[hc] + cc * We[HC + hc]);
      m = m > 0.0f ? m : NEG_SLOPE * m;
      al += m * att[hc];
    }
    float a = expf(al - ord2f(amax[(long long)d * H + h]));
    atomAddF(&denom[(long long)d * H + h], a);
    for (int c = 0; c < C; ++c) {
      int hc = h * C + c;
      atomAddF(&acc[(long long)d * HC + hc], xl[(long long)s * HC + hc] * a);
    }
  }
}

// ---------------- GATv2: node finalize (normalize + bias + relu) ----------------
__global__ void k_gat_fin(const float* __restrict__ acc, const float* __restrict__ denom,
                          const float* __restrict__ bias, float* __restrict__ xout,
                          int N, int H, int C) {
  int n = blockIdx.x * blockDim.x + threadIdx.x;
  if (n >= N) return;
  int HC = H * C;
  for (int hc = 0; hc < HC; ++hc) {
    int h = hc / C;
    float dn = denom[(long long)n * H + h];
    float v = (dn > 0.0f) ? (acc[(long long)n * HC + hc] / dn) : 0.0f;
    v += bias[hc];
    xout[(long long)n * HC + hc] = v > 0.0f ? v : 0.0f;
  }
}

// ---------------- fcn: [N,32]@[32,32]+b, relu ----------------
__global__ void k_fcn(const float* __restrict__ xin, const float* __restrict__ W,
                      const float* __restrict__ b, float* __restrict__ xout, int N) {
  int n = blockIdx.x * blockDim.x + threadIdx.x;
  if (n >= N) return;
  float v[32];
#pragma unroll
  for (int i = 0; i < 32; ++i) v[i] = xin[(long long)n * 32 + i];
  for (int j = 0; j < 32; ++j) {
    float acc = b[j];
#pragma unroll
    for (int i = 0; i < 32; ++i) acc += v[i] * W[i * 32 + j];
    xout[(long long)n * 32 + j] = acc > 0.0f ? acc : 0.0f;
  }
}

// ---------------- edge head: [E,66]@[66,32]+b -> @[32,2]+b via f32 WMMA ----------------
// Per wave: one 16-edge tile. A = ef[16,68] (K padded 66->68), B = W1[68,32].
// f32 WMMA 16x16x4: 17 K-steps x 2 column-halves.
__global__ __launch_bounds__(256)
void k_edge_head(const int* __restrict__ ei, const float* __restrict__ ew,
                 const float* __restrict__ ce, const float* __restrict__ xf,
                 const float* __restrict__ W1, const float* __restrict__ B1,
                 const float* __restrict__ W2, const float* __restrict__ B2,
                 float* __restrict__ out, long long E) {
  __shared__ float sW1[68 * 32];
  __shared__ float sB1[32];
  __shared__ float sW2[64];
  __shared__ float sB2[2];
  __shared__ float sEF[8][16 * 68];
  __shared__ float sEF1[8][16 * 32];

  const int tid = threadIdx.x;
  const int wave = tid >> 5;
  const int lane = tid & 31;
  const int half = lane >> 4;
  const int l16 = lane & 15;

  // stage weights (K rows 66,67 zero-padded)
  for (int i = tid; i < 68 * 32; i += 256) {
    int k = i >> 5;
    sW1[i] = (k < 66) ? W1[i] : 0.0f;
  }
  if (tid < 32) sB1[tid] = B1[tid];
  if (tid < 64) sW2[tid] = W2[tid];
  if (tid < 2) sB2[tid] = B2[tid];
  __syncthreads();

  const long long tile = (long long)blockIdx.x * 8 + wave;
  const long long e0 = tile * 16;
  float* myEF = &sEF[wave][0];
  float* myEF1 = &sEF1[wave][0];

  // stage ef tile [16 rows x 68 cols]: {ew, ce, xf[src] (32), xf[dst] (32), 0, 0}
  for (int idx = lane; idx < 16 * 68; idx += 32) {
    int r = idx / 68;
    int c = idx - r * 68;
    long long e = e0 + r;
    float v = 0.0f;
    if (e < E && c < 66) {
      if (c == 0) v = ew[e];
      else if (c == 1) v = ce[e];
      else if (c < 34) { int s = ei[e]; v = xf[(long long)s * 32 + (c - 2)]; }
      else { int d = ei[E + e]; v = xf[(long long)d * 32 + (c - 34)]; }
    }
    myEF[idx] = v;
  }
  __syncthreads();

  // --- WMMA f32 16x16x4 accumulation over K=68 ---
  // A frag: lanes 0-15 hold row l16 K={4k,4k+1}; lanes 16-31 hold K={4k+2,4k+3}
  // B frag: V0: lanes0-15 K=4k+0, lanes16-31 K=4k+2; V1: K=4k+1 / 4k+3
  v8f c0 = {};
  v8f c1 = {};
  for (int kk = 0; kk < 17; ++kk) {
    int ka = kk * 4 + half * 2;
    v2f a, b0, b1;
    a.x = myEF[l16 * 68 + ka + 0];
    a.y = myEF[l16 * 68 + ka + 1];
    b0.x = sW1[(ka + 0) * 32 + l16];
    b0.y = sW1[(ka + 1) * 32 + l16];
    b1.x = sW1[(ka + 0) * 32 + 16 + l16];
    b1.y = sW1[(ka + 1) * 32 + 16 + l16];
    c0 = __builtin_amdgcn_wmma_f32_16x16x4_f32(false, a, false, b0, (short)0, c0, false, false);
    c1 = __builtin_amdgcn_wmma_f32_16x16x4_f32(false, a, false, b1, (short)0, c1, false, false);
  }

  // store D fragments (+ bias) into LDS ef1[16,32]
#pragma unroll
  for (int r = 0; r < 8; ++r) {
    int row = half ? (r + 8) : r;
    myEF1[row * 32 + l16] = c0[r] + sB1[l16];
    myEF1[row * 32 + 16 + l16] = c1[r] + sB1[16 + l16];
  }
  __syncthreads();

  // second GEMM [16,32]@[32,2]: 32 lanes = 16 edges x 2 outputs
  {
    int row = lane >> 1;
    int j = lane & 1;
    long long e = e0 + row;
    if (e < E) {
      float acc = sB2[j];
#pragma unroll
      for (int k = 0; k < 32; ++k) acc += myEF1[row * 32 + k] * sW2[k * 2 + j];
      out[e * 2 + j] = acc;
    }
  }
}

// ============================== host side ==============================
extern "C" void kernel_launch(void* const* d_in, const int* in_sizes, int n_in,
                              void* d_out, int out_size, void* d_ws, size_t ws_size,
                              hipStream_t stream) {
  const float* x = (const float*)d_in[0];
  const int* ei = (const int*)d_in[1];
  const float* ew = (const float*)d_in[2];
  const float* ce = (const float*)d_in[3];
  const float* pna_deg = (const float*)d_in[4];
  const float* pna_We = (const float*)d_in[5];
  const float* pna_be = (const float*)d_in[6];
  const float* pna_Wpre = (const float*)d_in[7];
  const float* pna_bpre = (const float*)d_in[8];
  const float* pna_Wpost = (const float*)d_in[9];
  const float* pna_bpost = (const float*)d_in[10];
  const float* pna_Wlin = (const float*)d_in[11];
  const float* pna_blin = (const float*)d_in[12];
  const float* g1_Wl = (const float*)d_in[13];
  const float* g1_bl = (const float*)d_in[14];
  const float* g1_Wr = (const float*)d_in[15];
  const float* g1_br = (const float*)d_in[16];
  const float* g1_We = (const float*)d_in[17];
  const float* g1_att = (const float*)d_in[18];
  const float* g1_bias = (const float*)d_in[19];
  const float* g2_Wl = (const float*)d_in[20];
  const float* g2_bl = (const float*)d_in[21];
  const float* g2_Wr = (const float*)d_in[22];
  const float* g2_br = (const float*)d_in[23];
  const float* g2_We = (const float*)d_in[24];
  const float* g2_att = (const float*)d_in[25];
  const float* g2_bias = (const float*)d_in[26];
  const float* fcn_W = (const float*)d_in[27];
  const float* fcn_b = (const float*)d_in[28];
  const float* fce1_W = (const float*)d_in[29];
  const float* fce1_b = (const float*)d_in[30];
  const float* fce2_W = (const float*)d_in[31];
  const float* fce2_b = (const float*)d_in[32];
  float* out = (float*)d_out;

  const int N = in_sizes[0] / 5;
  const long long E = in_sizes[2];

  // workspace layout (floats)
  size_t o = 0;
  float* ws = (float*)d_ws;
  float* avg = ws;            o += 16;
  float* cnt = ws + o;        o += (size_t)N;          // zero-fill region starts here
  float* sum = ws + o;        o += (size_t)N * 5;
  float* sumsq = ws + o;      o += (size_t)N * 5;      // ...11N zeros
  unsigned* minb = (unsigned*)(ws + o); o += (size_t)N * 5;
  unsigned* maxb = (unsigned*)(ws + o); o += (size_t)N * 5;
  unsigned* amax = (unsigned*)(ws + o); o += (size_t)N * 4;  // GAT zero region start
  float* denom = ws + o;      o += (size_t)N * 4;
  float* accb = ws + o;       o += (size_t)N * 32;     // ...40N zeros per layer
  float* xl = ws + o;         o += (size_t)N * 32;
  float* xr = ws + o;         o += (size_t)N * 32;
  float* nodeA = ws + o;      o += (size_t)N * 32;
  float* nodeB = ws + o;      o += (size_t)N * 32;
  if (o * sizeof(float) > ws_size) return;  // insufficient scratch (deterministic no-op)

  const int TB = 256;
  const int nBlkN = (N + TB - 1) / TB;
  const long long nBlkE = (E + TB - 1) / TB;
  const int fillBlk = 2048;

  // --- init ---
  k_fill_u32<<<fillBlk, TB, 0, stream>>>((unsigned*)cnt, 0u, (size_t)N * 11);
  k_fill_u32<<<fillBlk, TB, 0, stream>>>(minb, 0xFFFFFFFFu, (size_t)N * 5);
  k_fill_u32<<<fillBlk, TB, 0, stream>>>(maxb, 0u, (size_t)N * 5);
  k_avglog<<<1, 64, 0, stream>>>(pna_deg, avg);

  // --- PNA ---
  k_pna_edge<<<(unsigned)nBlkE, TB, 0, stream>>>(ei, ew, ce, x, pna_We, pna_be, pna_Wpre,
                                                 pna_bpre, cnt, sum, sumsq, minb, maxb, E);
  k_pna_node<<<nBlkN, TB, 0, stream>>>(x, cnt, sum, sumsq, minb, maxb, avg, pna_Wpost,
                                       pna_bpost, pna_Wlin, pna_blin, nodeA, N);

  // --- GATv2 layer 1: in=8 -> H=4,C=4 (out 16), nodeA -> nodeB ---
  k_fill_u32<<<fillBlk, TB, 0, stream>>>(amax, 0u, (size_t)N * 40);
  k_gat_node<<<nBlkN, TB, 0, stream>>>(nodeA, 8, 16, g1_Wl, g1_bl, g1_Wr, g1_br, xl, xr, N);
  k_gat_edge_a<<<(unsigned)nBlkE, TB, 0, stream>>>(ei, ew, ce, xl, xr, g1_We, g1_att, amax, E, 4, 4);
  k_gat_edge_b<<<(unsigned)nBlkE, TB, 0, stream>>>(ei, ew, ce, xl, xr, g1_We, g1_att, amax,
                                                   denom, accb, E, 4, 4);
  k_gat_fin<<<nBlkN, TB, 0, stream>>>(accb, denom, g1_bias, nodeB, N, 4, 4);

  // --- GATv2 layer 2: in=16 -> H=4,C=8 (out 32), nodeB -> nodeA ---
  k_fill_u32<<<fillBlk, TB, 0, stream>>>(amax, 0u, (size_t)N * 40);
  k_gat_node<<<nBlkN, TB, 0, stream>>>(nodeB, 16, 32, g2_Wl, g2_bl, g2_Wr, g2_br, xl, xr, N);
  k_gat_edge_a<<<(unsigned)nBlkE, TB, 0, stream>>>(ei, ew, ce, xl, xr, g2_We, g2_att, amax, E, 4, 8);
  k_gat_edge_b<<<(unsigned)nBlkE, TB, 0, stream>>>(ei, ew, ce, xl, xr, g2_We, g2_att, amax,
                                                   denom, accb, E, 4, 8);
  k_gat_fin<<<nBlkN, TB, 0, stream>>>(accb, denom, g2_bias, nodeA, N, 4, 8);

  // --- fcn relu: nodeA -> nodeB ---
  k_fcn<<<nBlkN, TB, 0, stream>>>(nodeA, fcn_W, fcn_b, nodeB, N);

  // --- edge head with f32 WMMA ---
  long long tiles = (E + 15) / 16;
  long long headBlocks = (tiles + 7) / 8;
  k_edge_head<<<(unsigned)headBlocks, 256, 0, stream>>>(ei, ew, ce, nodeB, fce1_W, fce1_b,
                                                        fce2_W, fce2_b, out, E);
}